// InhibitoryNetwork_146028888488
// MI455X (gfx1250) — compile-verified
//
#include <hip/hip_runtime.h>

#define PYR 16384
#define PV  1638
#define SST 1228
#define VIP 819

// ---- workspace layout (float offsets) ----
#define WS_U      0
#define WS_PYRF   (PYR)
#define WS_PVF    (2*PYR)
#define WS_SSTF   (WS_PVF  + PV)
#define WS_VIPF   (WS_SSTF + SST)
#define WS_ACC    (WS_VIPF + VIP)      // start of zero-initialized accumulators
#define WS_PVEXC  (WS_ACC)
#define WS_PVINH  (WS_PVEXC  + PV)
#define WS_SSTEXC (WS_PVINH  + PV)
#define WS_SSTINH (WS_SSTEXC + SST)
#define WS_VIPEXC (WS_SSTINH + SST)
#define WS_VIPTMP (WS_VIPEXC + VIP)
#define WS_END    (WS_VIPTMP + PYR)

// ---- d_out layout (float offsets, reference tuple order) ----
#define O_TOT  0
#define O_PVS  (PYR)
#define O_SSTS (O_PVS  + PV)
#define O_VIPS (O_SSTS + SST)
#define O_PERI (O_VIPS + VIP)
#define O_DEND (O_PERI + PYR)

typedef float v2f __attribute__((ext_vector_type(2)));
typedef float v4f __attribute__((ext_vector_type(4)));
typedef float v8f __attribute__((ext_vector_type(8)));

__device__ __forceinline__ v8f wmma_f32(v2f a, v2f b, v8f c) {
  return __builtin_amdgcn_wmma_f32_16x16x4_f32(false, a, false, b,
                                               (short)0, c, false, false);
}

// ---------------------------------------------------------------------------
// Tensor Data Mover staging of the x-vector into LDS (one DMA per workgroup,
// issued by wave 0; completion via TENSORcnt + workgroup barrier).
// D# built per CDNA5 ISA 8.3/8.4: 1 x K row tile of 4-byte elements.
// ---------------------------------------------------------------------------
#ifndef __has_builtin
#define __has_builtin(x) 0
#endif
#if __has_builtin(__builtin_amdgcn_tensor_load_to_lds) && \
    __has_builtin(__builtin_amdgcn_s_wait_tensorcnt)
#define USE_TDM 1
#else
#define USE_TDM 0
#endif

#if USE_TDM
typedef unsigned int u32x4 __attribute__((ext_vector_type(4)));
typedef int          i32x4 __attribute__((ext_vector_type(4)));
typedef int          i32x8 __attribute__((ext_vector_type(8)));

__device__ __forceinline__ void tdm_load_vec(float* smem, const float* x, int K) {
  const unsigned long long ga = (unsigned long long)(uintptr_t)x;
  const unsigned int lds = (unsigned int)(uintptr_t)smem; // flat->LDS: addr[31:0]
  u32x4 g0;
  g0.x = 1u;                                   // count=1 (valid), user mode
  g0.y = lds;                                  // lds_addr (bytes)
  g0.z = (unsigned int)ga;                     // global_addr[31:0]
  g0.w = (unsigned int)((ga >> 32) & 0x01FFFFFFull) | (2u << 30); // [56:32] | type=2
  i32x8 g1;
  g1[0] = (2 << 16);                           // data_size = 4B; wg_mask=0
  g1[1] = (K & 0xFFFF) << 16;                  // tensor_dim0[15:0] @ bits 63:48
  g1[2] = ((K >> 16) & 0xFFFF) | (1 << 16);    // tensor_dim0[31:16] | tensor_dim1=1
  g1[3] = (K & 0xFFFF) << 16;                  // tile_dim0 = K @ bits 127:112
  g1[4] = 1;                                   // tile_dim1 = 1, tile_dim2 = 0
  g1[5] = K;                                   // tensor_dim0_stride[31:0] = K
  g1[6] = 0;
  g1[7] = 0;
  const i32x4 z4 = {0, 0, 0, 0};
#if __clang_major__ >= 23
  const i32x8 z8 = {0, 0, 0, 0, 0, 0, 0, 0};
  __builtin_amdgcn_tensor_load_to_lds(g0, g1, z4, z4, z8, 0);
#else
  __builtin_amdgcn_tensor_load_to_lds(g0, g1, z4, z4, 0);
#endif
}
#endif

__device__ __forceinline__ void stage_x(float* smem, const float* __restrict__ x,
                                        int K, int Kpad, int tid) {
#if USE_TDM
  if (tid < 32) {                       // wave 0: one TDM DMA for the whole vector
    tdm_load_vec(smem, x, K);
    for (int i = K + tid; i < Kpad; i += 32) smem[i] = 0.0f;  // zero tail pad
    __builtin_amdgcn_s_wait_tensorcnt(0);
  }
  __syncthreads();
#else
  for (int i = tid; i < Kpad; i += 256) smem[i] = (i < K) ? x[i] : 0.0f;
  __syncthreads();
#endif
}

// ---------------------------------------------------------------------------
// y[r] += yscale * sum_k W[r,k] * x[k]   (W row-major rows x K, fp32)
// One 16-row tile per wave, V_WMMA_F32_16X16X4_F32, x staged in LDS and fed
// unmasked to all lanes (only D column 0 is read; other columns are benign
// duplicate matvecs -> no exec-masking needed on B).
// AL: 4 -> K%4==0 (b128 A-loads + shfl_xor redistribution)
//     2 -> K even  (b64 A-loads, native A layout)
//     1 -> K odd   (b32 A-loads)
// ---------------------------------------------------------------------------
template <int AL>
__global__ void __launch_bounds__(256) matvec_wmma(
    const float* __restrict__ W, const float* __restrict__ x,
    float* __restrict__ y, int rows, int K, int Kpad, float yscale)
{
  extern __shared__ float smem[];
  const int tid = threadIdx.x;
  stage_x(smem, x, K, Kpad, tid);

  const int lane = tid & 31;
  const int wav  = tid >> 5;
  const int r0   = (blockIdx.x * 8 + wav) * 16;
  const int rit  = lane & 15;          // row within tile (A: lane%16 = M)
  const int hi   = lane >> 4;          // A layout: lanes 16-31 hold K+2 slots
  int r = r0 + rit; if (r > rows - 1) r = rows - 1;   // clamp partial tiles
  const float* __restrict__ Wr = W + (long long)r * (long long)K;

  v8f acc = {0.f, 0.f, 0.f, 0.f, 0.f, 0.f, 0.f, 0.f};
  int k0 = 0;

  if constexpr (AL == 4) {
    // 16B-aligned rows: lane(hi=0) loads row cols k0..k0+3, lane(hi=1) loads
    // cols k0+4..k0+7 of the SAME row; shfl_xor(16) swap -> two WMMA A tiles.
    const int Kmain = K & ~7;
    const float* __restrict__ p = Wr + 4 * hi;
    for (; k0 < Kmain; k0 += 8, p += 8) {
      v4f av = *(const v4f*)p;                 // global_load_b128
      v4f tv;
      tv.x = __shfl_xor(av.x, 16, 32);
      tv.y = __shfl_xor(av.y, 16, 32);
      tv.z = __shfl_xor(av.z, 16, 32);
      tv.w = __shfl_xor(av.w, 16, 32);
      v2f a1, a2;
      a1.x = hi ? tv.z : av.x;  a1.y = hi ? tv.w : av.y;   // k slots k0+2hi..+1
      a2.x = hi ? av.z : tv.x;  a2.y = hi ? av.w : tv.y;   // k slots k0+4+2hi..+1
      v2f b1 = *(const v2f*)(smem + k0 + 2 * hi);          // ds_load_b64
      v2f b2 = *(const v2f*)(smem + k0 + 4 + 2 * hi);
      acc = wmma_f32(a1, b1, acc);
      acc = wmma_f32(a2, b2, acc);
    }
  } else if constexpr (AL == 2) {
    // 8B-aligned rows: b64 A-load directly matches WMMA A layout.
    const int Kmain = K & ~3;
    const float* __restrict__ p = Wr + 2 * hi;
    for (; k0 < Kmain; k0 += 4, p += 4) {
      v2f a = *(const v2f*)p;                  // global_load_b64
      v2f b = *(const v2f*)(smem + k0 + 2 * hi);
      acc = wmma_f32(a, b, acc);
    }
  } else {
    // odd K: rows only 4B-aligned -> scalar loads.
    const int Kmain = K & ~3;
    const float* __restrict__ p = Wr + 2 * hi;
    for (; k0 < Kmain; k0 += 4, p += 4) {
      v2f a; a.x = p[0]; a.y = p[1];
      v2f b = *(const v2f*)(smem + k0 + 2 * hi);
      acc = wmma_f32(a, b, acc);
    }
  }

  // clamped tail tile(s): smem zero-padding nullifies slots k >= K
  for (; k0 < Kpad; k0 += 4) {
    const int kk = k0 + 2 * hi;
    const int k1 = (kk     < K) ? kk     : (K - 1);
    const int k2 = (kk + 1 < K) ? kk + 1 : (K - 1);
    v2f a; a.x = Wr[k1]; a.y = Wr[k2];
    v2f b = *(const v2f*)(smem + kk);
    acc = wmma_f32(a, b, acc);
  }

  // D column 0: lane 0 -> rows r0..r0+7, lane 16 -> rows r0+8..r0+15
  if (rit == 0) {
    const int mb = r0 + hi * 8;
#pragma unroll
    for (int j = 0; j < 8; ++j) {
      const int rr = mb + j;
      if (rr < rows) y[rr] += yscale * acc[j];
    }
  }
}

// ---------------------------------------------------------------------------
// Gap junctions: y[r] += yscale * ( (W @ vm)[r] - vm[r] * rowsum(W)[r] )
// B column 0 = vm (lanes rit==0), column 1 = ones (lanes rit==1):
// matvec and rowsum from a single WMMA pass. K = PV = 1638 (even).
// ---------------------------------------------------------------------------
__global__ void __launch_bounds__(256) matvec_gap_wmma(
    const float* __restrict__ W, const float* __restrict__ x,
    float* __restrict__ y, int rows, int K, int Kpad, float yscale)
{
  extern __shared__ float smem[];
  const int tid = threadIdx.x;
  stage_x(smem, x, K, Kpad, tid);

  const int lane = tid & 31;
  const int wav  = tid >> 5;
  const int r0   = (blockIdx.x * 8 + wav) * 16;
  const int rit  = lane & 15;
  const int hi   = lane >> 4;
  int r = r0 + rit; if (r > rows - 1) r = rows - 1;
  const float* __restrict__ Wr = W + (long long)r * (long long)K;
  const bool ones_lane = (rit == 1);

  v8f acc = {0.f, 0.f, 0.f, 0.f, 0.f, 0.f, 0.f, 0.f};
  int k0 = 0;
  const int Kmain = K & ~3;
  const float* __restrict__ p = Wr + 2 * hi;
  for (; k0 < Kmain; k0 += 4, p += 4) {
    v2f a = *(const v2f*)p;                    // K even -> 8B aligned
    v2f b = *(const v2f*)(smem + k0 + 2 * hi);
    if (ones_lane) { b.x = 1.0f; b.y = 1.0f; } // column 1 = rowsum
    acc = wmma_f32(a, b, acc);
  }
  for (; k0 < Kpad; k0 += 4) {
    const int kk = k0 + 2 * hi;
    const int k1 = (kk     < K) ? kk     : (K - 1);
    const int k2 = (kk + 1 < K) ? kk + 1 : (K - 1);
    v2f a; a.x = Wr[k1]; a.y = Wr[k2];
    v2f b = *(const v2f*)(smem + kk);
    if (ones_lane) {
      b.x = (kk     < K) ? 1.0f : 0.0f;
      b.y = (kk + 1 < K) ? 1.0f : 0.0f;
    }
    acc = wmma_f32(a, b, acc);
  }

  const int mb = r0 + hi * 8;
#pragma unroll
  for (int j = 0; j < 8; ++j) {
    const float rs = __shfl(acc[j], lane | 1, 32);  // rowsum from column-1 lane
    if (rit == 0) {
      const int rr = mb + j;
      if (rr < rows) y[rr] += yscale * (acc[j] - smem[rr] * rs);
    }
  }
}

// ---------------------------------------------------------------------------
// Prep: sum(ext) reduction, u = pyr_f + (0.3/(1+0.01*sum))*ext, bool->float
// conversions, zero accumulators. Single block; trivial vs 520MB streaming.
// ---------------------------------------------------------------------------
__global__ void __launch_bounds__(1024) prep_kernel(
    const int* __restrict__ pyr_spk, const float* __restrict__ ext,
    const int* __restrict__ pv_spk, const int* __restrict__ sst_spk,
    const int* __restrict__ vip_spk, float* __restrict__ ws,
    float* __restrict__ out)
{
  __shared__ float red[1024];
  const int tid = threadIdx.x;
  float s = 0.0f;
  for (int i = tid; i < PYR; i += 1024) s += ext[i];
  red[tid] = s;
  __syncthreads();
  for (int off = 512; off > 0; off >>= 1) {
    if (tid < off) red[tid] += red[tid + off];
    __syncthreads();
  }
  const float c = 0.3f / (1.0f + red[0] * 0.01f);

  for (int i = tid; i < PYR; i += 1024) {
    const float pf = pyr_spk[i] ? 1.0f : 0.0f;
    ws[WS_PYRF + i] = pf;
    ws[WS_U + i]    = pf + c * ext[i];
  }
  for (int i = tid; i < PV;  i += 1024) ws[WS_PVF  + i] = pv_spk[i]  ? 1.0f : 0.0f;
  for (int i = tid; i < SST; i += 1024) ws[WS_SSTF + i] = sst_spk[i] ? 1.0f : 0.0f;
  for (int i = tid; i < VIP; i += 1024) ws[WS_VIPF + i] = vip_spk[i] ? 1.0f : 0.0f;

  const int accN = WS_END - WS_ACC;
  for (int i = tid; i < accN;    i += 1024) ws[WS_ACC + i]  = 0.0f;
  for (int i = tid; i < 2 * PYR; i += 1024) out[O_PERI + i] = 0.0f;
}

// ---------------------------------------------------------------------------
__global__ void __launch_bounds__(256) lif_kernel(
    const float* __restrict__ ws, const float* __restrict__ v_pv,
    const float* __restrict__ v_sst, const float* __restrict__ v_vip,
    float* __restrict__ out)
{
  const int i = blockIdx.x * blockDim.x + threadIdx.x;
  if (i < PV) {
    const float e = ws[WS_PVEXC + i], h = ws[WS_PVINH + i], v = v_pv[i];
    const float vn = v + (-v + e * (3.0f - v) + h * (-0.5f - v)) * (1.0f / 5.0f);
    out[O_PVS + i] = (vn >= 0.8f) ? 1.0f : 0.0f;
  }
  if (i < SST) {
    const float e = ws[WS_SSTEXC + i], h = ws[WS_SSTINH + i], v = v_sst[i];
    const float vn = v + (-v + e * (3.0f - v) + h * (-0.5f - v)) * (1.0f / 15.0f);
    out[O_SSTS + i] = (vn >= 1.0f) ? 1.0f : 0.0f;
  }
  if (i < VIP) {
    const float e = ws[WS_VIPEXC + i], v = v_vip[i];
    const float vn = v + (-v + e * (3.0f - v)) * (1.0f / 10.0f);
    out[O_VIPS + i] = (vn >= 0.9f) ? 1.0f : 0.0f;
  }
}

__global__ void __launch_bounds__(256) finish_kernel(
    const float* __restrict__ ws, float* __restrict__ out)
{
  const int i = blockIdx.x * blockDim.x + threadIdx.x;
  if (i < PYR) out[O_TOT + i] = out[O_PERI + i] + out[O_DEND + i] + ws[WS_VIPTMP + i];
}

// ---------------------------------------------------------------------------
extern "C" void kernel_launch(void* const* d_in, const int* in_sizes, int n_in,
                              void* d_out, int out_size, void* d_ws, size_t ws_size,
                              hipStream_t stream) {
  (void)in_sizes; (void)n_in; (void)out_size; (void)ws_size;
  const int*   pyr_spk  = (const int*)  d_in[0];
  const float* ext      = (const float*)d_in[2];
  const int*   pv_spk   = (const int*)  d_in[3];
  const int*   sst_spk  = (const int*)  d_in[4];
  const int*   vip_spk  = (const int*)  d_in[5];
  const float* pv_mem   = (const float*)d_in[6];
  const float* v_pv     = (const float*)d_in[7];
  const float* v_sst    = (const float*)d_in[8];
  const float* v_vip    = (const float*)d_in[9];
  const float* w_pyr_pv  = (const float*)d_in[10];
  const float* w_pyr_sst = (const float*)d_in[11];
  const float* w_pyr_vip = (const float*)d_in[12];
  const float* w_pv_pyr  = (const float*)d_in[13];
  const float* w_sst_pyr = (const float*)d_in[14];
  const float* w_vip_pyr = (const float*)d_in[15];
  const float* w_pv_pv   = (const float*)d_in[16];
  const float* w_pv_sst  = (const float*)d_in[17];
  const float* w_sst_pv  = (const float*)d_in[18];
  const float* w_vip_pv  = (const float*)d_in[19];
  const float* w_vip_sst = (const float*)d_in[20];
  const float* w_pv_gap  = (const float*)d_in[21];

  float* ws  = (float*)d_ws;
  float* out = (float*)d_out;

  prep_kernel<<<1, 1024, 0, stream>>>(pyr_spk, ext, pv_spk, sst_spk, vip_spk, ws, out);

  auto mv = [&](const float* W, const float* x, float* y, int rows, int K, float sc) {
    const int Kpad = (K + 3) & ~3;
    const int grid = (rows + 127) / 128;
    const size_t lds = (size_t)Kpad * sizeof(float);
    if ((K & 3) == 0)
      matvec_wmma<4><<<grid, 256, lds, stream>>>(W, x, y, rows, K, Kpad, sc);
    else if ((K & 1) == 0)
      matvec_wmma<2><<<grid, 256, lds, stream>>>(W, x, y, rows, K, Kpad, sc);
    else
      matvec_wmma<1><<<grid, 256, lds, stream>>>(W, x, y, rows, K, Kpad, sc);
  };

  // E -> I excitation (K = 16384)
  mv(w_pyr_pv,  ws + WS_U,    ws + WS_PVEXC,  PV,  PYR, 1.0f);
  mv(w_pyr_sst, ws + WS_PYRF, ws + WS_SSTEXC, SST, PYR, 1.0f);
  mv(w_pyr_vip, ws + WS_PYRF, ws + WS_VIPEXC, VIP, PYR, 1.25f);  // 1 + ACH*0.5

  // I -> I inhibition (accumulating into zeroed buffers)
  mv(w_pv_pv,   ws + WS_PVF,  ws + WS_PVINH,  PV,  PV,  1.0f);
  mv(w_sst_pv,  ws + WS_SSTF, ws + WS_PVINH,  PV,  SST, 1.0f);
  mv(w_vip_pv,  ws + WS_VIPF, ws + WS_PVINH,  PV,  VIP, 1.0f);
  mv(w_pv_sst,  ws + WS_PVF,  ws + WS_SSTINH, SST, PV,  1.0f);
  mv(w_vip_sst, ws + WS_VIPF, ws + WS_SSTINH, SST, VIP, 1.0f);

  // gap junctions: pv_exc += 0.3 * (W@vm - vm * rowsum(W))
  {
    const int Kpad = (PV + 3) & ~3;
    matvec_gap_wmma<<<(PV + 127) / 128, 256, Kpad * sizeof(float), stream>>>(
        w_pv_gap, pv_mem, ws + WS_PVEXC, PV, PV, Kpad, 0.3f);
  }

  lif_kernel<<<(PV + 255) / 256, 256, 0, stream>>>(ws, v_pv, v_sst, v_vip, out);

  // I -> E feedback (rows = 16384)
  mv(w_pv_pyr,  out + O_PVS,  out + O_PERI,   PYR, PV,  1.0f);
  mv(w_sst_pyr, out + O_SSTS, out + O_DEND,   PYR, SST, 1.0f);
  mv(w_vip_pyr, out + O_VIPS, ws + WS_VIPTMP, PYR, VIP, 1.0f);

  finish_kernel<<<(PYR + 255) / 256, 256, 0, stream>>>(ws, out);
}